// SingleViewGNN_35682588295428
// MI455X (gfx1250) — compile-verified
//
#include <hip/hip_runtime.h>
#include <math.h>

typedef __attribute__((ext_vector_type(2))) float v2f;
typedef __attribute__((ext_vector_type(8))) float v8f;

#define NMI 1043
#define NDR 2166
#define HDIM 128

// ---------------------------------------------------------------- utilities
__global__ void set_val_kernel(float* __restrict__ p, int n, float v) {
  int i = blockIdx.x * blockDim.x + threadIdx.x;
  if (i < n) p[i] = v;
}

// Wt[c][r] = W[r][c], 128x128
__global__ void transpose128_kernel(const float* __restrict__ W, float* __restrict__ Wt) {
  int i = blockIdx.x * blockDim.x + threadIdx.x;
  if (i >= HDIM * HDIM) return;
  int r = i >> 7, c = i & (HDIM - 1);
  Wt[(size_t)c * HDIM + r] = W[i];
}

// ew[e] = attr[src[e]][dst[e]];  deg[dst[e]] += ew[e]
__global__ void edge_gather_deg_kernel(const int* __restrict__ edge, int E,
                                       const float* __restrict__ attr, int n,
                                       float* __restrict__ ew,
                                       float* __restrict__ deg) {
  int e = blockIdx.x * blockDim.x + threadIdx.x;
  if (e >= E) return;
  int s = edge[e];
  int d = edge[E + e];
  float w = attr[(size_t)s * n + d];
  ew[e] = w;
  __hip_atomic_fetch_add(&deg[d], w, __ATOMIC_RELAXED, __HIP_MEMORY_SCOPE_AGENT);
}

__global__ void dinv_kernel(const float* __restrict__ deg, float* __restrict__ dinv, int n) {
  int i = blockIdx.x * blockDim.x + threadIdx.x;
  if (i >= n) return;
  float d = deg[i];
  dinv[i] = (d > 0.f) ? rsqrtf(fmaxf(d, 1e-12f)) : 0.f;
}

// acc[i][c] = h[i][c] * dinv[i]^2   (self-loop contribution, coef = dinv*1*dinv)
__global__ void self_init_kernel(const float* __restrict__ h,
                                 const float* __restrict__ dinv,
                                 float* __restrict__ acc, int total) {
  int i = blockIdx.x * blockDim.x + threadIdx.x;
  if (i >= total) return;
  float dv = dinv[i >> 7];
  acc[i] = h[i] * dv * dv;
}

// one wave per edge; lane l handles channels [4l, 4l+3]
__global__ void gcn_scatter_kernel(const int* __restrict__ edge, int E,
                                   const float* __restrict__ ew,
                                   const float* __restrict__ dinv,
                                   const float* __restrict__ h,
                                   float* __restrict__ acc) {
  int gid = blockIdx.x * blockDim.x + threadIdx.x;
  int e = gid >> 5;
  int lane = gid & 31;
  if (e >= E) return;
  int s = edge[e];
  int d = edge[E + e];
  float coef = dinv[s] * ew[e] * dinv[d];
  const float4* hp = (const float4*)(h + (size_t)s * HDIM);
  float4 v = hp[lane];
  float* ap = acc + (size_t)d * HDIM + lane * 4;
  __hip_atomic_fetch_add(ap + 0, v.x * coef, __ATOMIC_RELAXED, __HIP_MEMORY_SCOPE_AGENT);
  __hip_atomic_fetch_add(ap + 1, v.y * coef, __ATOMIC_RELAXED, __HIP_MEMORY_SCOPE_AGENT);
  __hip_atomic_fetch_add(ap + 2, v.z * coef, __ATOMIC_RELAXED, __HIP_MEMORY_SCOPE_AGENT);
  __hip_atomic_fetch_add(ap + 3, v.w * coef, __ATOMIC_RELAXED, __HIP_MEMORY_SCOPE_AGENT);
}

__global__ void bias_relu_kernel(const float* __restrict__ acc, const float* __restrict__ b,
                                 float* __restrict__ x, int total) {
  int i = blockIdx.x * blockDim.x + threadIdx.x;
  if (i >= total) return;
  x[i] = fmaxf(acc[i] + b[i & (HDIM - 1)], 0.f);
}

__global__ void reduce_sum_kernel(const float* __restrict__ x, int n, float* __restrict__ out) {
  __shared__ float sdata[256];
  float acc = 0.f;
  for (int i = blockIdx.x * blockDim.x + threadIdx.x; i < n; i += gridDim.x * blockDim.x)
    acc += x[i];
  sdata[threadIdx.x] = acc;
  __syncthreads();
  for (int s = 128; s > 0; s >>= 1) {
    if ((int)threadIdx.x < s) sdata[threadIdx.x] += sdata[threadIdx.x + s];
    __syncthreads();
  }
  if (threadIdx.x == 0)
    __hip_atomic_fetch_add(out, sdata[0], __ATOMIC_RELAXED, __HIP_MEMORY_SCOPE_AGENT);
}

// att = relu(sigmoid(relu(mean @ fc1W.T + fc1b) @ fc2W.T + fc2b))
__global__ void attention_mlp_kernel(const float* __restrict__ sums, float inv_cnt,
                                     const float* __restrict__ fc1W, const float* __restrict__ fc1b,
                                     const float* __restrict__ fc2W, const float* __restrict__ fc2b,
                                     float* __restrict__ att) {
  float a0 = sums[0] * inv_cnt, a1 = sums[1] * inv_cnt;
  float t0 = fmaxf(a0 * fc1W[0] + a1 * fc1W[1] + fc1b[0], 0.f);
  float t1 = fmaxf(a0 * fc1W[2] + a1 * fc1W[3] + fc1b[1], 0.f);
  float u0 = t0 * fc2W[0] + t1 * fc2W[1] + fc2b[0];
  float u1 = t0 * fc2W[2] + t1 * fc2W[3] + fc2b[1];
  att[0] = fmaxf(1.f / (1.f + expf(-u0)), 0.f);
  att[1] = fmaxf(1.f / (1.f + expf(-u1)), 0.f);
}

// xcat[i][h] = relu(att[h<128?0:1] * x{1,2}[i][h%128]),   n x 256
__global__ void scale_concat_kernel(const float* __restrict__ x1, const float* __restrict__ x2,
                                    const float* __restrict__ att,
                                    float* __restrict__ xcat, int total) {
  int i = blockIdx.x * blockDim.x + threadIdx.x;
  if (i >= total) return;
  int node = i >> 8;
  int hh = i & 255;
  float v = (hh < 128) ? att[0] * x1[node * 128 + hh]
                       : att[1] * x2[node * 128 + (hh - 128)];
  xcat[i] = fmaxf(v, 0.f);
}

// ---------------------------------------------------------------- WMMA GEMM
// C[M x N] = A[M x K] * Bt^T + (bias), where Bt is [N x K] row-major
// (Bt[n][k] = B[k][n]) so all fragment loads are contiguous b64 loads.
//
// One wave32 computes a 32x32 C tile: 2 A-frags x 2 B-frags -> 4 accumulators,
// 4x V_WMMA_F32_16X16X4_F32 per k-step with full A/B fragment reuse.
// Out-of-range rows/cols are CLAMPED (not zero-filled): row m of A only feeds
// row m of C, col n of B only feeds col n of C, and those are never stored —
// this keeps the inner loop branch-free with EXEC constant (WMMA requirement).
//
// Frag layouts (ISA 7.12.2): A/B frag lane(lo)=row/col, v2f = k(2*hi, 2*hi+1).
// D: vgpr v -> row v + 8*hi, col = lane(lo).
template <bool BIAS>
__global__ void __launch_bounds__(32)
gemm_tn_wmma_kernel(const float* __restrict__ A, int lda,
                    const float* __restrict__ Bt, int ldb,
                    const float* __restrict__ bias,
                    float* __restrict__ C, int ldc,
                    int M, int N, int K) {
  const int tileM = blockIdx.x * 32;
  const int tileN = blockIdx.y * 32;
  const int lane = threadIdx.x;
  const int lo = lane & 15;
  const int hi = lane >> 4;

  int r0 = tileM + lo;       if (r0 >= M) r0 = M - 1;
  int r1 = tileM + 16 + lo;  if (r1 >= M) r1 = M - 1;
  int n0 = tileN + lo;       if (n0 >= N) n0 = N - 1;
  int n1 = tileN + 16 + lo;  if (n1 >= N) n1 = N - 1;

  const float* ap0 = A + (size_t)r0 * lda + 2 * hi;
  const float* ap1 = A + (size_t)r1 * lda + 2 * hi;
  const float* bp0 = Bt + (size_t)n0 * ldb + 2 * hi;
  const float* bp1 = Bt + (size_t)n1 * ldb + 2 * hi;

  v8f c00 = {}, c01 = {}, c10 = {}, c11 = {};
#pragma unroll 4
  for (int k0 = 0; k0 < K; k0 += 4) {
    v2f a0 = *(const v2f*)(ap0 + k0);
    v2f a1 = *(const v2f*)(ap1 + k0);
    v2f b0 = *(const v2f*)(bp0 + k0);
    v2f b1 = *(const v2f*)(bp1 + k0);
    c00 = __builtin_amdgcn_wmma_f32_16x16x4_f32(false, a0, false, b0, (short)0, c00, false, false);
    c01 = __builtin_amdgcn_wmma_f32_16x16x4_f32(false, a0, false, b1, (short)0, c01, false, false);
    c10 = __builtin_amdgcn_wmma_f32_16x16x4_f32(false, a1, false, b0, (short)0, c10, false, false);
    c11 = __builtin_amdgcn_wmma_f32_16x16x4_f32(false, a1, false, b1, (short)0, c11, false, false);
  }

  const int colL = tileN + lo;       // unclamped, for store guards
  const int colR = tileN + 16 + lo;
  const float bvL = (BIAS && colL < N) ? bias[colL] : 0.f;
  const float bvR = (BIAS && colR < N) ? bias[colR] : 0.f;
#pragma unroll
  for (int v = 0; v < 8; ++v) {
    const int rowT = tileM + v + 8 * hi;
    const int rowB = rowT + 16;
    if (rowT < M) {
      if (colL < N) C[(size_t)rowT * ldc + colL] = c00[v] + bvL;
      if (colR < N) C[(size_t)rowT * ldc + colR] = c01[v] + bvR;
    }
    if (rowB < M) {
      if (colL < N) C[(size_t)rowB * ldc + colL] = c10[v] + bvL;
      if (colR < N) C[(size_t)rowB * ldc + colR] = c11[v] + bvR;
    }
  }
}

// ---------------------------------------------------------------- host side
static inline int cdiv(int a, int b) { return (a + b - 1) / b; }

static void run_branch(const float* emb, const int* edge, int E, const float* attr, int n,
                       const float* W1, const float* b1, const float* W2, const float* b2,
                       const float* fc1W, const float* fc1b, const float* fc2W, const float* fc2b,
                       const float* convW, const float* convb, float* Mout,
                       float* ew, float* deg, float* dinv, float* h, float* acc,
                       float* x1, float* x2, float* xcat, float* Wt, float* scal,
                       hipStream_t stream) {
  const int total = n * HDIM;

  // degrees (init 1.0 for self-loop), edge weight gather
  set_val_kernel<<<cdiv(n, 256), 256, 0, stream>>>(deg, n, 1.0f);
  set_val_kernel<<<1, 32, 0, stream>>>(scal, 2, 0.0f);
  edge_gather_deg_kernel<<<cdiv(E, 256), 256, 0, stream>>>(edge, E, attr, n, ew, deg);
  dinv_kernel<<<cdiv(n, 256), 256, 0, stream>>>(deg, dinv, n);

  dim3 gM(cdiv(n, 32), cdiv(HDIM, 32)), blk(32);
  const int tgrid = cdiv(HDIM * HDIM, 256);

  // GCN layer 1: h = emb @ W1 ; acc = selfloop + scatter ; x1 = relu(acc + b1)
  transpose128_kernel<<<tgrid, 256, 0, stream>>>(W1, Wt);
  gemm_tn_wmma_kernel<false><<<gM, blk, 0, stream>>>(
      emb, HDIM, Wt, HDIM, nullptr, h, HDIM, n, HDIM, HDIM);
  self_init_kernel<<<cdiv(total, 256), 256, 0, stream>>>(h, dinv, acc, total);
  gcn_scatter_kernel<<<cdiv(E * 32, 128), 128, 0, stream>>>(edge, E, ew, dinv, h, acc);
  bias_relu_kernel<<<cdiv(total, 256), 256, 0, stream>>>(acc, b1, x1, total);

  // GCN layer 2
  transpose128_kernel<<<tgrid, 256, 0, stream>>>(W2, Wt);
  gemm_tn_wmma_kernel<false><<<gM, blk, 0, stream>>>(
      x1, HDIM, Wt, HDIM, nullptr, h, HDIM, n, HDIM, HDIM);
  self_init_kernel<<<cdiv(total, 256), 256, 0, stream>>>(h, dinv, acc, total);
  gcn_scatter_kernel<<<cdiv(E * 32, 128), 128, 0, stream>>>(edge, E, ew, dinv, h, acc);
  bias_relu_kernel<<<cdiv(total, 256), 256, 0, stream>>>(acc, b2, x2, total);

  // channel attention (global mean pool -> 2x2 MLP -> sigmoid)
  reduce_sum_kernel<<<256, 256, 0, stream>>>(x1, total, scal + 0);
  reduce_sum_kernel<<<256, 256, 0, stream>>>(x2, total, scal + 1);
  attention_mlp_kernel<<<1, 1, 0, stream>>>(scal, 1.0f / (float)total,
                                            fc1W, fc1b, fc2W, fc2b, scal + 2);

  // conv fusion: Mout = [att0*x1 | att1*x2] (n x 256) @ convW^T (256 x 128) + convb
  // convW flat [o][c][h] = convW[o*256 + (c*128+h)] is already the Bt layout.
  scale_concat_kernel<<<cdiv(n * 256, 256), 256, 0, stream>>>(x1, x2, scal + 2, xcat, n * 256);
  gemm_tn_wmma_kernel<true><<<gM, blk, 0, stream>>>(
      xcat, 2 * HDIM, convW, 2 * HDIM, convb, Mout, HDIM, n, HDIM, 2 * HDIM);
}

extern "C" void kernel_launch(void* const* d_in, const int* in_sizes, int n_in,
                              void* d_out, int out_size, void* d_ws, size_t ws_size,
                              hipStream_t stream) {
  const float* mirna  = (const float*)d_in[0];
  const float* drug   = (const float*)d_in[1];
  const int*   mmE    = (const int*)d_in[2];
  const float* mmAttr = (const float*)d_in[3];
  const int*   ddE    = (const int*)d_in[4];
  const float* ddAttr = (const float*)d_in[5];
  const float* mW1 = (const float*)d_in[6];   const float* mb1 = (const float*)d_in[7];
  const float* mW2 = (const float*)d_in[8];   const float* mb2 = (const float*)d_in[9];
  const float* dW1 = (const float*)d_in[10];  const float* db1 = (const float*)d_in[11];
  const float* dW2 = (const float*)d_in[12];  const float* db2 = (const float*)d_in[13];
  const float* m_fc1W = (const float*)d_in[14]; const float* m_fc1b = (const float*)d_in[15];
  const float* m_fc2W = (const float*)d_in[16]; const float* m_fc2b = (const float*)d_in[17];
  const float* d_fc1W = (const float*)d_in[18]; const float* d_fc1b = (const float*)d_in[19];
  const float* d_fc2W = (const float*)d_in[20]; const float* d_fc2b = (const float*)d_in[21];
  const float* m_convW = (const float*)d_in[22]; const float* m_convb = (const float*)d_in[23];
  const float* d_convW = (const float*)d_in[24]; const float* d_convb = (const float*)d_in[25];

  const int E_mm = in_sizes[2] / 2;
  const int E_dd = in_sizes[4] / 2;
  const int E_max = (E_mm > E_dd) ? E_mm : E_dd;

  // workspace layout (floats, all region sizes multiple of 4 for float4 align)
  float* ws = (float*)d_ws;
  size_t o = 0;
  auto carve = [&](size_t nf) { float* p = ws + o; o += (nf + 3) & ~(size_t)3; return p; };
  float* MoutM = carve((size_t)NMI * HDIM);
  float* MoutD = carve((size_t)NDR * HDIM);
  float* ew    = carve((size_t)E_max);
  float* deg   = carve(NDR);
  float* dinv  = carve(NDR);
  float* h     = carve((size_t)NDR * HDIM);
  float* acc   = carve((size_t)NDR * HDIM);
  float* x1    = carve((size_t)NDR * HDIM);
  float* x2    = carve((size_t)NDR * HDIM);
  float* xcat  = carve((size_t)NDR * 2 * HDIM);
  float* Wt    = carve((size_t)HDIM * HDIM);
  float* scal  = carve(8);  // [0..1]=pool sums, [2..3]=attention scalars

  run_branch(mirna, mmE, E_mm, mmAttr, NMI,
             mW1, mb1, mW2, mb2, m_fc1W, m_fc1b, m_fc2W, m_fc2b,
             m_convW, m_convb, MoutM,
             ew, deg, dinv, h, acc, x1, x2, xcat, Wt, scal, stream);

  run_branch(drug, ddE, E_dd, ddAttr, NDR,
             dW1, db1, dW2, db2, d_fc1W, d_fc1b, d_fc2W, d_fc2b,
             d_convW, d_convb, MoutD,
             ew, deg, dinv, h, acc, x1, x2, xcat, Wt, scal, stream);

  // out[1043 x 2166] = MoutM @ MoutD^T  (MoutD is already Bt-layout [N x K])
  dim3 gf(cdiv(NMI, 32), cdiv(NDR, 32));
  gemm_tn_wmma_kernel<false><<<gf, dim3(32), 0, stream>>>(
      MoutM, HDIM, MoutD, HDIM, nullptr, (float*)d_out, NDR, NMI, NDR, HDIM);
}